// LDTW_29068338659749
// MI455X (gfx1250) — compile-verified
//
#include <hip/hip_runtime.h>

typedef __attribute__((ext_vector_type(2))) float v2f;
typedef __attribute__((ext_vector_type(8))) float v8f;

#define BATCH 16
#define NN 256
#define DIMK 64
#define BIGV 1000000000.0f
#define THRESHV 100000000.0f
#define BANDW 51            // floor(0.2*256); |i-j|<=51 allowed
#define NDIAG 103           // 2*BANDW+1
#define PITCH 106           // NDIAG + 2 halo cols + pad
#define ROWS 257            // rows 0..256 (row 0 = halo/seed)
#define BUFSZ (ROWS * PITCH)
#define NCELL (NN * NDIAG)  // 26368 live-band rectangle cells
#define NT 1024
#define MAXC ((NCELL + NT - 1) / NT)        // 26 cells per thread
#define CORNER ((NN - 1) * NDIAG + BANDW)   // cell (i=256, d=0) -> 26316

// ---------------- row norms: nX[b][i] = ||X_b[i]||^2 -----------------------
__global__ __launch_bounds__(256) void ldtw_norms(const float* __restrict__ X,
                                                  const float* __restrict__ Y,
                                                  float* __restrict__ nX,
                                                  float* __restrict__ nY) {
  int idx = blockIdx.x * blockDim.x + threadIdx.x;  // 2*16*256 = 8192
  if (idx >= 2 * BATCH * NN) return;
  int which = idx >> 12;      // 0 -> X, 1 -> Y (4096 rows each)
  int row = idx & 4095;
  const float* p = (which ? Y : X) + (size_t)row * DIMK;
  float s = 0.0f;
#pragma unroll
  for (int k = 0; k < DIMK; ++k) s += p[k] * p[k];
  (which ? nY : nX)[row] = s;
}

// ---------------- D[b][i][j] = nX + nY - 2 * (X Y^T) via f32 WMMA ----------
// One wave per 16x16 output tile. A = X tile (16x4), B = Y^T tile (4x16).
__global__ __launch_bounds__(256) void ldtw_dist_wmma(const float* __restrict__ X,
                                                      const float* __restrict__ Y,
                                                      const float* __restrict__ nX,
                                                      const float* __restrict__ nY,
                                                      float* __restrict__ D) {
  int wave = blockIdx.x * (blockDim.x >> 5) + (threadIdx.x >> 5);  // 0..4095
  int lane = threadIdx.x & 31;
  int b = wave >> 8;          // 256 tiles per batch
  int tile = wave & 255;
  int tm = (tile >> 4) << 4;
  int tn = (tile & 15) << 4;
  const float* Xb = X + (size_t)b * NN * DIMK;
  const float* Yb = Y + (size_t)b * NN * DIMK;
  int laneLo = lane & 15;     // M (for A) / N (for B)
  int laneHi = lane >> 4;     // K sub-select

  v8f acc = {};
  const float* arow = Xb + (size_t)(tm + laneLo) * DIMK;
  const float* brow = Yb + (size_t)(tn + laneLo) * DIMK;
#pragma unroll
  for (int k0 = 0; k0 < DIMK; k0 += 4) {
    // A 16x4 f32 layout: lanes 0-15 hold K=0,1 ; lanes 16-31 hold K=2,3
    v2f a = *(const v2f*)(arow + k0 + 2 * laneHi);
    // B 4x16 f32 layout: VGPR0 lanes 0-15 K=0 / 16-31 K=1 ; VGPR1 K=2 / K=3
    v2f bt;
    bt.x = brow[k0 + laneHi];
    bt.y = brow[k0 + 2 + laneHi];
    acc = __builtin_amdgcn_wmma_f32_16x16x4_f32(false, a, false, bt,
                                                (short)0, acc, false, false);
  }

  float ny = nY[b * NN + tn + laneLo];
  float* Dt = D + (size_t)b * NN * NN;
#pragma unroll
  for (int r = 0; r < 8; ++r) {
    int M = r + 8 * laneHi;   // C/D layout: VGPR r -> M=r (lanes 0-15), M=r+8 (lanes 16-31)
    int Ncol = laneLo;
    float nx = nX[b * NN + tm + M];
    Dt[(size_t)(tm + M) * NN + (tn + Ncol)] = nx + ny - 2.0f * acc[r];
  }
}

// ---------------- band-limited length-constrained DTW DP in LDS ------------
// Diagonal-skewed storage: cell (i, d=j-i) -> lds[i*PITCH + (d+BANDW+1)].
// Halo cols 0 & PITCH-2.. and row 0 stay BIG forever (except the (0,0) seed).
__global__ __launch_bounds__(NT, 1) void ldtw_dp(const float* __restrict__ D,
                                                 float* __restrict__ out) {
  extern __shared__ float lds[];
  float* bufA = lds;
  float* bufB = lds + BUFSZ;
  const int b = blockIdx.x;
  const int tid = threadIdx.x;

  for (int idx = tid; idx < 2 * BUFSZ; idx += NT) lds[idx] = BIGV;
  __syncthreads();
  if (tid == 0) bufA[BANDW + 1] = 0.0f;   // prev0[(0,0)] = 0 (row 0, d=0)

  // Per-thread static cell assignment; D values cached in registers.
  const float* Db = D + (size_t)b * NN * NN;
  float Dv[MAXC];
  int adr[MAXC];
  unsigned vm = 0u;
#pragma unroll
  for (int c = 0; c < MAXC; ++c) {
    int cell = tid + c * NT;
    int i = cell / NDIAG + 1;           // 1..256
    int dIdx = cell % NDIAG;            // d = dIdx - BANDW
    int j = i + dIdx - BANDW;
    bool ok = (cell < NCELL) && (j >= 1) && (j <= NN);
    adr[c] = i * PITCH + dIdx + 1;
    Dv[c] = ok ? Db[(size_t)(i - 1) * NN + (j - 1)] : BIGV;
    if (ok) vm |= (1u << c);
  }
  __syncthreads();

  float cmin = BIGV;
#pragma unroll 1
  for (int k = 0; k < 512; ++k) {
    const float* src = (k & 1) ? bufB : bufA;
    float* dst = (k & 1) ? bufA : bufB;
#pragma unroll
    for (int c = 0; c < MAXC; ++c) {
      if ((vm >> c) & 1u) {
        int a = adr[c];
        // prev[i-1][j-1] -> (i-1, d)   : a - PITCH
        // prev[i-1][j]   -> (i-1, d+1) : a - PITCH + 1
        // prev[i][j-1]   -> (i,   d-1) : a - 1
        float m = fminf(fminf(src[a - PITCH], src[a - PITCH + 1]), src[a - 1]);
        float val = (m < THRESHV) ? (m + Dv[c]) : BIGV;
        dst[a] = val;
        if (c == CORNER / NT) {                 // compile-time prune to c==25
          if (tid == (CORNER % NT) && k >= 255) cmin = fminf(cmin, val);
        }
      }
    }
    // bufA is reused as dst at k==1: retire the (0,0)=0 seed so k>=2 reads BIG.
    if (k == 1 && tid == 0) dst[BANDW + 1] = BIGV;
    __syncthreads();
  }
  if (tid == (CORNER % NT)) out[b] = cmin;
}

extern "C" void kernel_launch(void* const* d_in, const int* in_sizes, int n_in,
                              void* d_out, int out_size, void* d_ws, size_t ws_size,
                              hipStream_t stream) {
  const float* X = (const float*)d_in[0];
  const float* Y = (const float*)d_in[1];
  float* D = (float*)d_ws;                         // 16*256*256 f32 = 4 MB
  float* nX = D + (size_t)BATCH * NN * NN;         // 4096 f32
  float* nY = nX + (size_t)BATCH * NN;             // 4096 f32
  float* out = (float*)d_out;                      // 16 f32

  ldtw_norms<<<(2 * BATCH * NN + 255) / 256, 256, 0, stream>>>(X, Y, nX, nY);
  // 16 batches * 256 tiles = 4096 waves; 8 waves (256 thr) per block -> 512 blocks
  ldtw_dist_wmma<<<512, 256, 0, stream>>>(X, Y, nX, nY, D);
  // one workgroup per batch; 218 KB dynamic LDS (fits CDNA5's 320 KB WGP LDS)
  ldtw_dp<<<BATCH, NT, 2 * BUFSZ * sizeof(float), stream>>>(D, out);
}